// SoftDTWBatch_71073118814387
// MI455X (gfx1250) — compile-verified
//
#include <hip/hip_runtime.h>
#include <cstdint>
#include <cstddef>

// Soft-DTW forward on gfx1250 (MI455X).
//
// Dependency-depth-bound problem: N+M-1 = 1023 wavefront steps. Strategy:
//   - 128 workgroups (one per batch) x 512 threads (16 wave32) -> fills WGPs.
//   - Thread ti owns DP row i=ti+1; at step d it computes R(i, j=d-i).
//       left = own register; {up, diag} = neighbor ti-1's last two values,
//       packed float2 in LDS  ->  1x ds_load_b64 + 1x ds_store_b64 per step.
//   - Hand-coded split workgroup barrier per step; async-stage issue,
//     s_wait_asynccnt, next-step ring read and prefetch are placed INSIDE the
//     signal->wait window so they overlap barrier latency.
//   - D streaming: per-lane-gather global_load_async_to_lds_b32 into a
//     per-wave LDS ring, pipeline depth 8, retired in-order (ASYNCcnt).
//   - Whole DP runs scaled by log2(e): v_exp_f32/v_log_f32 are native base-2,
//     so softmin needs zero extra multiplies; one *ln2 at the end.

#define SD_B 128
#define SD_N 512
#define SD_M 512
#define LOOK 8          // async pipeline depth (slots in flight)
#define RINGD 16        // ring slots per wave (2x LOOK, power of two)

#define LOG2E 1.44269504088896340736f
#define LN2   0.69314718055994530942f

// softmin in the base-2-scaled domain: inputs are R*log2e, output is
// (-log(e^-a+e^-b+e^-c))*log2e computed as  m2 - log2(sum 2^(m2-x2)).
__device__ __forceinline__ float softmin3_2(float a2, float b2, float c2) {
    float m2 = fminf(a2, fminf(b2, c2));
    float s = __builtin_amdgcn_exp2f(m2 - a2)   // v_exp_f32; 2^(m-inf)=0
            + __builtin_amdgcn_exp2f(m2 - b2)
            + __builtin_amdgcn_exp2f(m2 - c2);
    return m2 - __builtin_amdgcn_logf(s);       // v_log_f32 (base-2)
}

__launch_bounds__(SD_N, 1)
__global__ void softdtw_fwd_kernel(const float* __restrict__ D,
                                   float* __restrict__ out) {
    const int b    = blockIdx.x;
    const int ti   = threadIdx.x;        // row-1: 0..N-1
    const int lane = ti & 31;
    const int wv   = ti >> 5;            // wave id 0..15
    const float INF = __builtin_inff();

    // pbuf[p][t] = (scaled R of thread t at step parity p, value one step older)
    __shared__ float2 pbuf[2][SD_N];                 // 8 KB
    __shared__ float  ring[SD_N / 32][RINGD][32];    // 32 KB, per-wave D stage

    const float* Drow = D + (size_t)b * (SD_N * SD_M) + (size_t)ti * SD_M;

    // LDS byte offset of this lane's ring column (generic-pointer truncation).
    const unsigned ring_base = (unsigned)(uintptr_t)(&ring[wv][0][lane]);

    // ---- prologue: stage D for steps d = 2 .. 2+LOOK-1 (ASYNCcnt -> LOOK) ----
    #pragma unroll
    for (int p = 0; p < LOOK; ++p) {
        int dp = 2 + p;
        int c  = dp - 2 - ti;
        c = (c < 0) ? 0 : ((c > SD_M - 1) ? (SD_M - 1) : c);
        unsigned lds_off = ring_base + (unsigned)((dp & (RINGD - 1)) * 32 * 4);
        unsigned long long ga = (unsigned long long)(uintptr_t)(Drow + c);
        asm volatile("global_load_async_to_lds_b32 %0, %1, off"
                     :: "v"(lds_off), "v"(ga) : "memory");
    }

    float myPrev = INF;     // scaled R(i, j-1); INF until first active step
    float result = 0.0f;

    // Pre-read D value for step d=2 (oldest of the 8 outstanding ops).
    asm volatile("s_wait_asynccnt 7" ::: "memory");
    float dval = ring[wv][2 & (RINGD - 1)][lane];

    #pragma unroll 1
    for (int d = 2; d <= SD_N + SD_M; ++d) {
        const int i = ti + 1;
        const int j = d - i;
        const bool active = (j >= 1) && (j <= SD_M);

        // ---------------- critical-path section ----------------
        if (active) {
            float up, diag;
            if (i == 1) {
                up   = INF;
                diag = (j == 1) ? 0.0f : INF;     // R(0,0)=0, R(0,j>=1)=inf
            } else {
                float2 pr = pbuf[(d - 1) & 1][ti - 1];  // ds_load_b64
                up   = pr.x;                            // R(i-1, j)
                diag = (j == 1) ? INF : pr.y;           // R(i-1, j-1)
            }
            float left = myPrev;                        // R(i, j-1)

            float cur = fmaf(dval, LOG2E, softmin3_2(diag, up, left));
            pbuf[d & 1][ti] = make_float2(cur, myPrev); // ds_store_b64
            myPrev = cur;
            if (d == SD_N + SD_M && ti == SD_N - 1) result = cur; // R(N,M)*log2e
        }

        // Make pbuf store visible, then arrive at the workgroup barrier.
        asm volatile("s_wait_dscnt 0x0\n\t"
                     "s_barrier_signal -1" ::: "memory");

        // -------- overlap window (no pbuf deps): hide barrier latency --------
        {   // stage D for step d+LOOK (always, clamped -> ASYNCcnt stays LOOK)
            int dn = d + LOOK;
            int c  = dn - 2 - ti;
            c = (c < 0) ? 0 : ((c > SD_M - 1) ? (SD_M - 1) : c);
            unsigned lds_off = ring_base + (unsigned)((dn & (RINGD - 1)) * 32 * 4);
            unsigned long long ga = (unsigned long long)(uintptr_t)(Drow + c);
            asm volatile("global_load_async_to_lds_b32 %0, %1, off"
                         :: "v"(lds_off), "v"(ga) : "memory");
        }
        // Retire the op for step d+1 (oldest of the 8 now outstanding) and
        // read next step's D value (lane-private slot; safe pre-barrier).
        asm volatile("s_wait_asynccnt 7" ::: "memory");
        dval = ring[wv][(d + 1) & (RINGD - 1)][lane];

        if ((d & 31) == 0) {            // warm L2 ~128 columns ahead
            int cp = d - 2 - ti + 128;
            cp = (cp < 0) ? 0 : ((cp > SD_M - 1) ? (SD_M - 1) : cp);
            __builtin_prefetch(Drow + cp, 0, 3);   // global_prefetch_b8
        }

        // Complete the barrier.
        asm volatile("s_barrier_wait -1" ::: "memory");
    }

    if (ti == SD_N - 1) out[b] = result * LN2;   // back to natural-log scale

    asm volatile("s_wait_asynccnt 0" ::: "memory");  // drain before endpgm
}

extern "C" void kernel_launch(void* const* d_in, const int* in_sizes, int n_in,
                              void* d_out, int out_size, void* d_ws, size_t ws_size,
                              hipStream_t stream) {
    (void)in_sizes; (void)n_in; (void)out_size; (void)d_ws; (void)ws_size;
    const float* D  = (const float*)d_in[0];   // (B, N, M) float32
    float* out      = (float*)d_out;           // (B,) float32
    softdtw_fwd_kernel<<<dim3(SD_B), dim3(SD_N), 0, stream>>>(D, out);
}